// GraphStructuralRNNConv_34780645163651
// MI455X (gfx1250) — compile-verified
//
#include <hip/hip_runtime.h>
#include <hip/hip_bf16.h>
#include <math.h>

// ---------------- problem constants (match reference) ----------------
constexpr int N_NODES = 50000;
constexpr int E_EDGES = 800000;
constexpr int D_FEAT  = 200;   // K == NC == 200
constexpr int B_BLK   = 100;   // 2x2 blocks per feature vector
constexpr int NT      = 13;    // ceil(200/16) column tiles
constexpr int ROW_TILES = N_NODES / 16;   // 3125 exactly

typedef float v2f __attribute__((ext_vector_type(2)));
typedef float v4f __attribute__((ext_vector_type(4)));
typedef float v8f __attribute__((ext_vector_type(8)));

// ---------------- utility: zero fp32 buffer (vectorized, n4 = count/4) ----------------
__global__ void zero_f32x4_kernel(v4f* __restrict__ p, int n4) {
    int i = blockIdx.x * blockDim.x + threadIdx.x;
    if (i < n4) p[i] = (v4f)0.0f;
}

// ---------------- degree accumulation (native integer atomics) ----------------
__global__ void degree_kernel(const int* __restrict__ dst, int* __restrict__ deg) {
    int e = blockIdx.x * blockDim.x + threadIdx.x;
    if (e < E_EDGES) atomicAdd(&deg[dst[e]], 1);
}

// int degree -> float node_norm, in place: deg>0 ? 1/deg : 0 (deg>=1 so max(d,1)==d)
__global__ void norm_kernel(void* __restrict__ buf) {
    int i = blockIdx.x * blockDim.x + threadIdx.x;
    if (i < N_NODES) {
        int c = ((const int*)buf)[i];
        ((float*)buf)[i] = (c > 0) ? (1.0f / (float)c) : 0.0f;
    }
}

// ---------------- BDD edge message + scatter-add ----------------
// One thread per (edge, block-pair): blocks 2p and 2p+1 (4 outputs).
// msg[e,b,o] = sum_i x[src[e], b*2+i] * w[et[e],b,i,o], scaled by norm[dst[e]],
// accumulated via native global_atomic_add_f32 (agg is L2-resident: 40 MB << 192 MB).
__global__ void edge_bdd_kernel(const float* __restrict__ x,
                                const int*   __restrict__ src,
                                const int*   __restrict__ dst,
                                const int*   __restrict__ et,
                                const float* __restrict__ w,     // (R,B,2,2)
                                const float* __restrict__ nrm,   // (N,)
                                float*       __restrict__ agg)   // (N,D)
{
    constexpr int PAIRS = B_BLK / 2;   // 50 block-pairs per edge
    int tid = blockIdx.x * blockDim.x + threadIdx.x;
    if (tid >= E_EDGES * PAIRS) return;
    int e = tid / PAIRS;
    int p = tid - e * PAIRS;           // pair index: covers blocks 2p, 2p+1

    int s = src[e];
    int d = dst[e];
    int r = et[e];
    float scale = nrm[d];

    // 4 consecutive inputs (16B aligned) -- coalesced across the 50 pair-threads
    v4f xin = *(const v4f*)(x + s * D_FEAT + p * 4);
    const float* wp = w + (r * B_BLK + 2 * p) * 4;
    v4f w0 = *(const v4f*)(wp);        // block 2p:   [i0o0, i0o1, i1o0, i1o1]
    v4f w1 = *(const v4f*)(wp + 4);    // block 2p+1

    float o0 = (xin.x * w0.x + xin.y * w0.z) * scale;
    float o1 = (xin.x * w0.y + xin.y * w0.w) * scale;
    float o2 = (xin.z * w1.x + xin.w * w1.z) * scale;
    float o3 = (xin.z * w1.y + xin.w * w1.w) * scale;

    float* ap = agg + d * D_FEAT + p * 4;
    unsafeAtomicAdd(ap + 0, o0);       // native global_atomic_add_f32
    unsafeAtomicAdd(ap + 1, o1);
    unsafeAtomicAdd(ap + 2, o2);
    unsafeAtomicAdd(ap + 3, o3);
}

// ---------------- WMMA fragment helper ----------------
// B (4x16) fragment from LDS-resident W (row-major, stride 200 words):
// lane<16 col=colC, v2f={K=kb,kb+1}; lane>=16 {K=kb+2,kb+3}.
// lanes16-31 sit +400 words = bank+16 -> conflict-free with lanes0-15.
__device__ __forceinline__ v2f load_b_lds(const float* Wl, int kA, int colC) {
    v2f b;
    b.x = Wl[kA * D_FEAT + colC];
    b.y = Wl[(kA + 1) * D_FEAT + colC];
    return b;
}

// ---------------- dense self-loop GEMM: out = op(addv + A @ W + bias) ----------------
// 8 waves = 8 row tiles per workgroup; W staged once in LDS (160 KB of the
// 320 KB/WGP budget) via float4 copies; each wave holds all 13 column-tile
// accumulators (104 VGPRs) and loads its A fragment exactly once per k-step.
// Garbage in cols 200..207 is never stored, so no zero-fill anywhere.
template<bool DOTANH>
__global__ void __launch_bounds__(256)
dense_wmma_kernel(const float* __restrict__ A,
                  const float* __restrict__ W,      // (200,200) row-major
                  const float* __restrict__ addv,   // (N,200)
                  const float* __restrict__ bias,   // (200,)
                  float*       __restrict__ out)    // (N,200)
{
    __shared__ float Wl[D_FEAT * D_FEAT];   // 160,000 B
    {
        v4f* dst4 = (v4f*)Wl;
        const v4f* src4 = (const v4f*)W;
        for (int i = threadIdx.x; i < (D_FEAT * D_FEAT) / 4; i += 256)
            dst4[i] = src4[i];              // coalesced b128 copy
    }
    __syncthreads();

    const int lane = threadIdx.x & 31;
    const int wv   = threadIdx.x >> 5;
    const int rowTile = blockIdx.x * 8 + wv;
    if (rowTile >= ROW_TILES) return;       // wave-uniform, after the only barrier

    const int half = lane >> 4;             // 0 or 1
    const int l16  = lane & 15;
    const float* Arow = A + (rowTile * 16 + l16) * D_FEAT;

    v8f acc[NT];
    #pragma unroll
    for (int t = 0; t < NT; ++t) acc[t] = (v8f)0.0f;

    for (int kb = 0; kb < D_FEAT; kb += 4) {
        const int kA = kb + 2 * half;
        v2f a = *(const v2f*)(Arow + kA);   // one global v2f per k-step
        #pragma unroll
        for (int t = 0; t < NT; ++t) {
            const int colC = min(t * 16 + l16, D_FEAT - 1);
            v2f b = load_b_lds(Wl, kA, colC);
            acc[t] = __builtin_amdgcn_wmma_f32_16x16x4_f32(
                         false, a, false, b, (short)0, acc[t], false, false);
        }
    }

    const int rowBase = rowTile * 16 + 8 * half;
    #pragma unroll
    for (int t = 0; t < NT; ++t) {
        const int col = t * 16 + l16;
        if (col < D_FEAT) {                 // per-lane store guard only
            float bs = bias[col];
            #pragma unroll
            for (int r = 0; r < 8; ++r) {
                int idx = (rowBase + r) * D_FEAT + col;
                float v = acc[t][r] + bs + addv[idx];
                if (DOTANH) v = tanhf(v);
                out[idx] = v;
            }
        }
    }
}

// ---------------- transpose-stage W^T into LDS, division-free ----------------
// wave wv handles k-rows wv, wv+8, ...; lanes cover columns (consecutive LDS
// words -> bank-conflict-free writes). Global reads are strided but the 160 KB
// matrix is L2/WGP$-hot and this runs once per phase.
__device__ __forceinline__ void stage_transposed(float* Wl, const float* W,
                                                 int wv, int lane) {
    for (int k = wv; k < D_FEAT; k += 8)
        for (int c = lane; c < D_FEAT; c += 32)
            Wl[k * D_FEAT + c] = W[c * D_FEAT + k];
}

// ---------------- RNN cell: out = tanh(A1 @ W1^T + b1 + A2 @ W2^T + b2) ----------------
// Stage W_ih^T, GEMM phase 1; barrier; restage W_hh^T, GEMM phase 2 into the
// same accumulators. Tail waves stay live through all three barriers.
__global__ void __launch_bounds__(256)
rnn_wmma_kernel(const float* __restrict__ A1, const float* __restrict__ W1, // W_ih
                const float* __restrict__ A2, const float* __restrict__ W2, // W_hh
                const float* __restrict__ b1, const float* __restrict__ b2,
                float* __restrict__ out)
{
    __shared__ float Wl[D_FEAT * D_FEAT];
    const int lane = threadIdx.x & 31;
    const int wv   = threadIdx.x >> 5;
    const int rowTile = blockIdx.x * 8 + wv;
    const bool active = rowTile < ROW_TILES;
    const int half = lane >> 4;
    const int l16  = lane & 15;

    v8f acc[NT];
    #pragma unroll
    for (int t = 0; t < NT; ++t) acc[t] = (v8f)0.0f;

    // ---- phase 1: W_ih^T ----
    stage_transposed(Wl, W1, wv, lane);
    __syncthreads();
    if (active) {
        const float* Arow = A1 + (rowTile * 16 + l16) * D_FEAT;
        for (int kb = 0; kb < D_FEAT; kb += 4) {
            const int kA = kb + 2 * half;
            v2f a = *(const v2f*)(Arow + kA);
            #pragma unroll
            for (int t = 0; t < NT; ++t) {
                const int colC = min(t * 16 + l16, D_FEAT - 1);
                v2f b = load_b_lds(Wl, kA, colC);
                acc[t] = __builtin_amdgcn_wmma_f32_16x16x4_f32(
                             false, a, false, b, (short)0, acc[t], false, false);
            }
        }
    }
    __syncthreads();

    // ---- phase 2: W_hh^T ----
    stage_transposed(Wl, W2, wv, lane);
    __syncthreads();
    if (active) {
        const float* Arow = A2 + (rowTile * 16 + l16) * D_FEAT;
        for (int kb = 0; kb < D_FEAT; kb += 4) {
            const int kA = kb + 2 * half;
            v2f a = *(const v2f*)(Arow + kA);
            #pragma unroll
            for (int t = 0; t < NT; ++t) {
                const int colC = min(t * 16 + l16, D_FEAT - 1);
                v2f b = load_b_lds(Wl, kA, colC);
                acc[t] = __builtin_amdgcn_wmma_f32_16x16x4_f32(
                             false, a, false, b, (short)0, acc[t], false, false);
            }
        }

        const int rowBase = rowTile * 16 + 8 * half;
        #pragma unroll
        for (int t = 0; t < NT; ++t) {
            const int col = t * 16 + l16;
            if (col < D_FEAT) {
                float bs = b1[col] + b2[col];
                #pragma unroll
                for (int r = 0; r < 8; ++r) {
                    int idx = (rowBase + r) * D_FEAT + col;
                    out[idx] = tanhf(acc[t][r] + bs);
                }
            }
        }
    }
}

// ---------------- host-side orchestration ----------------
extern "C" void kernel_launch(void* const* d_in, const int* in_sizes, int n_in,
                              void* d_out, int out_size, void* d_ws, size_t ws_size,
                              hipStream_t stream) {
    const float* node_feat   = (const float*)d_in[0];
    const float* dynamic_emb = (const float*)d_in[1];   // (N,1,D) -> flat N*D
    const int*   src         = (const int*)  d_in[2];
    const int*   dst         = (const int*)  d_in[3];
    const int*   etypes      = (const int*)  d_in[4];
    const float* w1          = (const float*)d_in[5];
    const float* loop_w1     = (const float*)d_in[6];
    const float* b1          = (const float*)d_in[7];
    const float* w2          = (const float*)d_in[8];
    const float* loop_w2     = (const float*)d_in[9];
    const float* b2          = (const float*)d_in[10];
    const float* W_ih        = (const float*)d_in[11];
    const float* W_hh        = (const float*)d_in[12];
    const float* b_ih        = (const float*)d_in[13];
    const float* b_hh        = (const float*)d_in[14];
    float* outp = (float*)d_out;

    // workspace layout (floats): norm | agg | h1 | h2
    float* ws   = (float*)d_ws;
    float* nrm  = ws;
    float* agg  = nrm + N_NODES;
    float* h1   = agg + (size_t)N_NODES * D_FEAT;
    float* h2   = h1  + (size_t)N_NODES * D_FEAT;

    const int TB = 256;
    const int nd = N_NODES * D_FEAT;                             // 10,000,000
    const int edgeThreads = E_EDGES * (B_BLK / 2);               // 40M
    const int denseBlocks = (ROW_TILES + 7) / 8;                 // 391

    // 1) zero norm + agg (count divisible by 4)
    {
        int n4 = (N_NODES + nd) / 4;
        zero_f32x4_kernel<<<(n4 + TB - 1) / TB, TB, 0, stream>>>((v4f*)nrm, n4);
    }
    // 2) degrees (int atomics) -> norm (float, in place)
    degree_kernel<<<(E_EDGES + TB - 1) / TB, TB, 0, stream>>>(dst, (int*)nrm);
    norm_kernel<<<(N_NODES + TB - 1) / TB, TB, 0, stream>>>(nrm);

    // 3) layer-1 edge messages
    edge_bdd_kernel<<<(edgeThreads + TB - 1) / TB, TB, 0, stream>>>(
        node_feat, src, dst, etypes, w1, nrm, agg);

    // 4) layer-1 dense: h1 = tanh(agg + node_feat @ loop_w1 + b1)
    dense_wmma_kernel<true><<<denseBlocks, TB, 0, stream>>>(
        node_feat, loop_w1, agg, b1, h1);

    // 5) re-zero agg for layer 2
    {
        int n4 = nd / 4;
        zero_f32x4_kernel<<<(n4 + TB - 1) / TB, TB, 0, stream>>>((v4f*)agg, n4);
    }
    // 6) layer-2 edge messages (input h1)
    edge_bdd_kernel<<<(edgeThreads + TB - 1) / TB, TB, 0, stream>>>(
        h1, src, dst, etypes, w2, nrm, agg);

    // 7) layer-2 dense (no tanh): h2 = agg + h1 @ loop_w2 + b2
    dense_wmma_kernel<false><<<denseBlocks, TB, 0, stream>>>(
        h1, loop_w2, agg, b2, h2);

    // 8) RNN cell: out = tanh(h2 @ W_ih^T + b_ih + h0 @ W_hh^T + b_hh)
    rnn_wmma_kernel<<<denseBlocks, TB, 0, stream>>>(
        h2, W_ih, dynamic_emb, W_hh, b_ih, b_hh, outp);
}